// PatchTokenizer_30425548325193
// MI455X (gfx1250) — compile-verified
//
#include <hip/hip_runtime.h>
#include <hip/hip_bf16.h>

// ---------------- problem constants ----------------
#define BB    32
#define CC    3
#define IMGSZ 512
#define GB0   32
#define GB1   16
#define GB2   8
#define N0    (BB * GB0 * GB0) /* 32768 base patches   */
#define N1    (BB * GB1 * GB1) /*  8192 scale-1 cells  */
#define N2    (BB * GB2 * GB2) /*  2048 scale-2 cells  */
#define TOKB  1345             /* 1 + 1024 + 256 + 64  */
#define NTOK  (BB * TOKB)      /* 43040 tokens         */

// ---- CDNA5 TDM availability ----
#if defined(__has_builtin)
#  if __has_builtin(__builtin_amdgcn_tensor_load_to_lds) && __has_builtin(__builtin_amdgcn_s_wait_tensorcnt)
#    define PT_HAVE_TDM 1
#  endif
#endif
#ifndef PT_HAVE_TDM
#  define PT_HAVE_TDM 0
#endif

typedef unsigned int u32x4 __attribute__((ext_vector_type(4)));
typedef int          i32x4 __attribute__((ext_vector_type(4)));
typedef int          i32x8 __attribute__((ext_vector_type(8)));

// =====================================================================
// F1: fulls[1] — per selected scale-2 cell: a 3x64x64 f32 tile
//     (48 KB, strides 512 / 512^2).  One TDM descriptor per block
//     (tensor_load_to_lds, TENSORcnt) pulls the whole strided tile
//     into LDS; waves then do the (c,row,col) -> (patch, c, 16x16)
//     permute with ds_load_b128 -> coalesced global_store_b128.
//     Fallback: per-lane global_load_async_to_lds_b128 (ASYNCcnt).
// =====================================================================
__global__ void pt_f1(const float* __restrict__ img, const int* __restrict__ idx,
                      const int* __restrict__ cntp, float* __restrict__ out) {
  __shared__ float tile[12288];  // (c,row,col) = 3 x 64 x 64 f32 = 48 KB
  const int r = blockIdx.x, t = threadIdx.x;  // 256 threads
  float4* out4 = reinterpret_cast<float4*>(out) + (size_t)r * 3072;
  if (r < *cntp) {
    const int j = idx[r];
    const int b = j >> 6, rr = j & 63, h2 = rr >> 3, w2 = rr & 7;
    const unsigned long long gaddr = (unsigned long long)(size_t)(
        img + (((size_t)(b * CC)) * IMGSZ + h2 * 64) * IMGSZ + w2 * 64);
#if PT_HAVE_TDM
    if (t == 0) {
      const unsigned ldsa = (unsigned)(size_t)&tile[0];
      // D# group0: count=1 | lds_addr | global_addr[56:0] | type=2
      u32x4 g0 = {1u, ldsa, (unsigned)(gaddr & 0xFFFFFFFFu),
                  ((unsigned)((gaddr >> 32) & 0x01FFFFFFu)) | 0x80000000u};
      // D# group1: data_size=4B; tensor_dim0/1=512; tile_dim={64,64,3};
      //            dim0_stride=512; dim1_stride=512*512 (=0x40000)
      i32x8 g1 = {0x00020000,             // wg_mask=0, data_size code 2 (4B)
                  (int)(512u << 16),      // tensor_dim0[15:0] @ bits 63:48
                  (int)(512u << 16),      // tensor_dim1[15:0] @ bits 95:80
                  (int)(64u << 16),       // tile_dim0 = 64 @ bits 127:112
                  (int)(64u | (3u << 16)),// tile_dim1=64, tile_dim2=3
                  512,                    // tensor_dim0_stride[31:0]
                  0,                      // stride0 hi | stride1[15:0]
                  4};                     // tensor_dim1_stride[47:16] = 0x4
      i32x4 g2 = {3, 0, 0, 0};            // tensor_dim2 = 3 (channels)
      i32x4 g3 = {0, 0, 0, 0};
      i32x8 g4 = {0, 0, 0, 0, 0, 0, 0, 0};// extra group (clang-23 6-arg form)
      __builtin_amdgcn_tensor_load_to_lds(g0, g1, g2, g3, g4, 0);
      __builtin_amdgcn_s_wait_tensorcnt(0);
    }
    asm volatile("" ::: "memory");
    __syncthreads();
#else
    // async fallback into the same (c,row,col) LDS layout
#pragma unroll 4
    for (int q = 0; q < 12; ++q) {
      const int e4 = t + q * 256;                 // float4 index in tile
      const int c = e4 >> 10, rem = e4 & 1023;    // 64*16 float4 per channel
      const int row = rem >> 4, col4 = rem & 15;
      const float* src = img +
          (((size_t)(b * CC + c)) * IMGSZ + h2 * 64 + row) * IMGSZ + w2 * 64 + col4 * 4;
      const unsigned lds_off = (unsigned)(size_t)&tile[e4 * 4];
      const unsigned long long ga2 = (unsigned long long)(size_t)src;
      asm volatile("global_load_async_to_lds_b128 %0, %1, off"
                   :: "v"(lds_off), "v"(ga2) : "memory");
    }
    asm volatile("s_wait_asynccnt 0" ::: "memory");
    __syncthreads();
#endif
    // drain: permute (c, row, col) -> (n=ny*4+nx, c, y, x), b128 stores
#pragma unroll 4
    for (int q = 0; q < 12; ++q) {
      const int e4 = t + q * 256;  // 0..3071 dest float4
      const int n = e4 / 192, rem = e4 % 192, c = rem >> 6, r2 = rem & 63;
      const int y = r2 >> 2, x4 = r2 & 3, ny = n >> 2, nx = n & 3;
      const int lidx = (c * 64 + ny * 16 + y) * 16 + nx * 4 + x4;
      out4[e4] = reinterpret_cast<const float4*>(tile)[lidx];
    }
  } else {
    const float4 z = make_float4(0.f, 0.f, 0.f, 0.f);
#pragma unroll 4
    for (int q = 0; q < 12; ++q) out4[t + q * 256] = z;
  }
}

// =====================================================================
// F0: fulls[0] — per selected scale-1 cell, its 2x2 base patches.
//     Async global->LDS staging (no VGPR return payload -> deep
//     pipelining), then coalesced b128 drain.  Row = 3072 floats.
// =====================================================================
__global__ void pt_f0(const float* __restrict__ img, const int* __restrict__ idx,
                      const int* __restrict__ cntp, float* __restrict__ out) {
  __shared__ float tile[3072];  // 12 KB, output-ordered
  const int r = blockIdx.x, t = threadIdx.x;  // 256 threads
  float4* out4 = reinterpret_cast<float4*>(out) + (size_t)r * 768;
  if (r < *cntp) {
    const int j = idx[r];
    const int b = j >> 8, rr = j & 255, h1 = rr >> 4, w1 = rr & 15;
#pragma unroll
    for (int q = 0; q < 3; ++q) {
      const int e4 = t + q * 256;  // 0..767 (output order)
      const int n = e4 / 192, rem = e4 % 192, c = rem >> 6, r2 = rem & 63;
      const int y = r2 >> 2, x4 = r2 & 3, ny = n >> 1, nx = n & 1;
      const int row = h1 * 32 + ny * 16 + y, col = w1 * 32 + nx * 16 + x4 * 4;
      const float* src = img + (((size_t)(b * CC + c)) * IMGSZ + row) * IMGSZ + col;
      const unsigned lds_off = (unsigned)(size_t)&tile[e4 * 4];
      const unsigned long long ga = (unsigned long long)(size_t)src;
      asm volatile("global_load_async_to_lds_b128 %0, %1, off"
                   :: "v"(lds_off), "v"(ga) : "memory");
    }
    asm volatile("s_wait_asynccnt 0" ::: "memory");
    __syncthreads();
#pragma unroll
    for (int q = 0; q < 3; ++q) {
      const int e4 = t + q * 256;
      out4[e4] = reinterpret_cast<const float4*>(tile)[e4];
    }
  } else {
    const float4 z = make_float4(0.f, 0.f, 0.f, 0.f);
#pragma unroll
    for (int q = 0; q < 3; ++q) out4[t + q * 256] = z;
  }
}

// =====================================================================
// G0: resized[0] — compacted base 16x16 patches (direct b128 gather).
// =====================================================================
__global__ void pt_g0(const float* __restrict__ img, const int* __restrict__ idx,
                      const int* __restrict__ cntp, float* __restrict__ out) {
  const int r = blockIdx.x, t = threadIdx.x;  // 192 threads
  float4 v = make_float4(0.f, 0.f, 0.f, 0.f);
  if (r < *cntp) {
    const int j = idx[r];
    const int b = j >> 10, rr = j & 1023, h = rr >> 5, w = rr & 31;
    const int c = t / 64, rem = t & 63, y = rem >> 2, x4 = rem & 3;
    const size_t src = (((size_t)(b * CC + c)) * IMGSZ + (h * 16 + y)) * IMGSZ + w * 16 + x4 * 4;
    v = *reinterpret_cast<const float4*>(img + src);
  }
  reinterpret_cast<float4*>(out)[(size_t)r * 192 + t] = v;
}

// =====================================================================
// G1: resized[1] — 512->256 bilinear is an exact 2x2 mean; fused.
// =====================================================================
__global__ void pt_g1(const float* __restrict__ img, const int* __restrict__ idx,
                      const int* __restrict__ cntp, float* __restrict__ out) {
  const int r = blockIdx.x, t = threadIdx.x;
  float4 v = make_float4(0.f, 0.f, 0.f, 0.f);
  if (r < *cntp) {
    const int j = idx[r];
    const int b = j >> 8, rr = j & 255, h1 = rr >> 4, w1 = rr & 15;
    const int c = t / 64, rem = t & 63, y = rem >> 2, x4 = rem & 3;
    const int Y = h1 * 16 + y;         // row in 256-res image
    const int col = w1 * 32 + x4 * 8;  // 8 source cols, 32B aligned
    const float* r0 = img + (((size_t)(b * CC + c)) * IMGSZ + 2 * Y) * IMGSZ + col;
    const float* r1 = r0 + IMGSZ;
    const float4 a0 = reinterpret_cast<const float4*>(r0)[0];
    const float4 a1 = reinterpret_cast<const float4*>(r0)[1];
    const float4 b0 = reinterpret_cast<const float4*>(r1)[0];
    const float4 b1 = reinterpret_cast<const float4*>(r1)[1];
    v.x = 0.25f * (a0.x + a0.y + b0.x + b0.y);
    v.y = 0.25f * (a0.z + a0.w + b0.z + b0.w);
    v.z = 0.25f * (a1.x + a1.y + b1.x + b1.y);
    v.w = 0.25f * (a1.z + a1.w + b1.z + b1.w);
  }
  reinterpret_cast<float4*>(out)[(size_t)r * 192 + t] = v;
}

// =====================================================================
// G2: resized[2] — 512->128 bilinear = mean of center 2x2 of each 4x4
//     block (rows/cols 4i+1,4i+2); fused.
// =====================================================================
__global__ void pt_g2(const float* __restrict__ img, const int* __restrict__ idx,
                      const int* __restrict__ cntp, float* __restrict__ out) {
  const int r = blockIdx.x, t = threadIdx.x;
  float4 v = make_float4(0.f, 0.f, 0.f, 0.f);
  if (r < *cntp) {
    const int j = idx[r];
    const int b = j >> 6, rr = j & 63, h2 = rr >> 3, w2 = rr & 7;
    const int c = t / 64, rem = t & 63, y = rem >> 2, x4 = rem & 3;
    const int Y = h2 * 16 + y;  // row in 128-res image
    const float* r0 = img + (((size_t)(b * CC + c)) * IMGSZ + (4 * Y + 1)) * IMGSZ;
    const float* r1 = r0 + IMGSZ;
    float o[4];
#pragma unroll
    for (int k = 0; k < 4; ++k) {
      const int X = w2 * 16 + x4 * 4 + k;
      const int cc = 4 * X + 1;
      o[k] = 0.25f * (r0[cc] + r0[cc + 1] + r1[cc] + r1[cc + 1]);
    }
    v = make_float4(o[0], o[1], o[2], o[3]);
  }
  reinterpret_cast<float4*>(out)[(size_t)r * 192 + t] = v;
}

// =====================================================================
// Fused control kernel: masks + seqlens + 3 stable scans + token
// compaction + cu_seqlens/max/frac.  ONE block, 1024 threads.
// =====================================================================
__device__ __forceinline__ float pt_tok_val(int p, const int* m0, const int* m1, const int* m2) {
  const int b = p / TOKB, q = p - b * TOKB;
  if (q == 0) return -1.0f;
  if (q < 1025) return m0[b * 1024 + (q - 1)] ? 1.0f : 0.0f;
  if (q < 1281) return m1[b * 256 + (q - 1025)] ? 2.0f : 0.0f;
  return m2[b * 64 + (q - 1281)] ? 3.0f : 0.0f;
}

__device__ __forceinline__ int pt_block_scan_compact(const int* __restrict__ flags, int n,
                                                     int* __restrict__ idxmap, int* s) {
  const int t = threadIdx.x;
  const int ipt = (n + 1023) >> 10;
  const int lo = t * ipt;
  const int hi = (lo + ipt < n) ? (lo + ipt) : n;
  int sum = 0;
  for (int i = lo; i < hi; ++i) sum += flags[i];
  __syncthreads();  // protect s[] reuse from previous phase
  s[t] = sum;
  __syncthreads();
  for (int off = 1; off < 1024; off <<= 1) {
    int v = (t >= off) ? s[t - off] : 0;
    __syncthreads();
    if (t >= off) s[t] += v;
    __syncthreads();
  }
  int run = s[t] - sum;
  for (int i = lo; i < hi; ++i)
    if (flags[i]) idxmap[run++] = i;
  const int total = s[1023];
  __syncthreads();
  return total;
}

__global__ void __launch_bounds__(1024)
pt_control(const float* __restrict__ imp1, const float* __restrict__ imp2,
           int* __restrict__ m0w, int* __restrict__ m1w, int* __restrict__ m2w,
           int* __restrict__ idx0, int* __restrict__ idx1, int* __restrict__ idx2,
           int* __restrict__ cnts,
           float* __restrict__ om0, float* __restrict__ om1, float* __restrict__ om2,
           float* __restrict__ ocnt, float* __restrict__ otok,
           float* __restrict__ oseq, float* __restrict__ ocu,
           float* __restrict__ omax, float* __restrict__ ofrac) {
  __shared__ int s[1024];
  __shared__ int sm2[64];
  __shared__ int sm1[256];
  __shared__ int seqs[32];
  const int t = threadIdx.x;

  // ---- phase A: hierarchical masks ----
  for (int b = 0; b < BB; ++b) {
    if (t < 64) sm2[t] = (imp2[b * 64 + t] < 0.3f) ? 1 : 0;
    __syncthreads();
    if (t < 256) {
      const int h1 = t >> 4, w1 = t & 15;
      sm1[t] = ((imp1[b * 256 + t] < 0.5f) && !sm2[(h1 >> 1) * 8 + (w1 >> 1)]) ? 1 : 0;
    }
    __syncthreads();
    const int h0 = t >> 5, w0 = t & 31;
    const int m0 = 1 ^ (sm2[(h0 >> 2) * 8 + (w0 >> 2)] | sm1[(h0 >> 1) * 16 + (w0 >> 1)]);
    m0w[b * 1024 + t] = m0;
    om0[b * 1024 + t] = (float)m0;
    if (t < 256) { m1w[b * 256 + t] = sm1[t]; om1[b * 256 + t] = (float)sm1[t]; }
    if (t < 64)  { m2w[b * 64 + t]  = sm2[t]; om2[b * 64 + t]  = (float)sm2[t]; }
    __syncthreads();  // protect sm1/sm2 before next iteration overwrites
  }

  // ---- phase B: per-batch seqlens ----
  if (t < 32) {
    int acc = 1;
    for (int i = 0; i < 1024; ++i) acc += m0w[t * 1024 + i];
    for (int i = 0; i < 256;  ++i) acc += m1w[t * 256 + i];
    for (int i = 0; i < 64;   ++i) acc += m2w[t * 64 + i];
    seqs[t] = acc;
    oseq[t] = (float)acc;
  }

  // ---- phase C: stable scans -> compaction index maps + counts ----
  int tot;
  tot = pt_block_scan_compact(m0w, N0, idx0, s);
  if (t == 0) { cnts[0] = tot; ocnt[0] = (float)tot; }
  tot = pt_block_scan_compact(m1w, N1, idx1, s);
  if (t == 0) { cnts[1] = tot; ocnt[1] = (float)tot; }
  tot = pt_block_scan_compact(m2w, N2, idx2, s);
  if (t == 0) { cnts[2] = tot; ocnt[2] = (float)tot; }

  // ---- phase D: token compaction (output_mask) ----
  {
    const int ipt = (NTOK + 1023) >> 10;  // 43
    const int lo = t * ipt;
    const int hi = (lo + ipt < NTOK) ? (lo + ipt) : NTOK;
    for (int i = lo; i < hi; ++i) otok[i] = 0.0f;
    int sum = 0;
    for (int i = lo; i < hi; ++i) sum += (pt_tok_val(i, m0w, m1w, m2w) != 0.0f) ? 1 : 0;
    __syncthreads();
    s[t] = sum;
    __syncthreads();
    for (int off = 1; off < 1024; off <<= 1) {
      int v = (t >= off) ? s[t - off] : 0;
      __syncthreads();
      if (t >= off) s[t] += v;
      __syncthreads();
    }
    int run = s[t] - sum;
    for (int i = lo; i < hi; ++i) {
      const float v = pt_tok_val(i, m0w, m1w, m2w);
      if (v != 0.0f) otok[run++] = v;
    }
  }
  __syncthreads();

  // ---- phase E: cu_seqlens / max / retained_frac ----
  if (t == 0) {
    int acc = 0, mx = 0;
    ocu[0] = 0.0f;
    for (int b = 0; b < 32; ++b) {
      const int sv = seqs[b];
      acc += sv;
      if (sv > mx) mx = sv;
      ocu[b + 1] = (float)acc;
    }
    *omax = (float)mx;
    *ofrac = (float)acc / 32768.0f;
  }
}

// =====================================================================
// host launcher
// =====================================================================
extern "C" void kernel_launch(void* const* d_in, const int* in_sizes, int n_in,
                              void* d_out, int out_size, void* d_ws, size_t ws_size,
                              hipStream_t stream) {
  (void)in_sizes; (void)n_in; (void)out_size; (void)ws_size;
  const float* img  = (const float*)d_in[0];  // (32,3,512,512)
  const float* imp1 = (const float*)d_in[2];  // (32,16,16)
  const float* imp2 = (const float*)d_in[3];  // (32,8,8)
  float* out = (float*)d_out;

  // ---- workspace carve (ints; ~344 KB) ----
  int* w    = (int*)d_ws;
  int* m0w  = w;          // 32768
  int* m1w  = m0w + N0;   // 8192
  int* m2w  = m1w + N1;   // 2048
  int* idx0 = m2w + N2;   // 32768
  int* idx1 = idx0 + N0;  // 8192
  int* idx2 = idx1 + N1;  // 2048
  int* cnts = idx2 + N2;  // 3

  // ---- output layout (floats, reference tuple order) ----
  const size_t o0  = 0;                        // resized[0]  32768*768
  const size_t o1  = o0 + (size_t)N0 * 768;    // resized[1]   8192*768
  const size_t o2  = o1 + (size_t)N1 * 768;    // resized[2]   2048*768
  const size_t o3  = o2 + (size_t)N2 * 768;    // fulls[0]     8192*3072
  const size_t o4  = o3 + (size_t)N1 * 3072;   // fulls[1]     2048*12288
  const size_t o5  = o4 + (size_t)N2 * 12288;  // mask0        32768
  const size_t o6  = o5 + N0;                  // mask1        8192
  const size_t o7  = o6 + N1;                  // mask2        2048
  const size_t o8  = o7 + N2;                  // counts       3
  const size_t o9  = o8 + 3;                   // output_mask  43040
  const size_t o10 = o9 + NTOK;                // seqlens      32
  const size_t o11 = o10 + 32;                 // cu_seqlens   33
  const size_t o12 = o11 + 33;                 // max          1
  const size_t o13 = o12 + 1;                  // retained     1

  // one fused control kernel (single block)
  pt_control<<<1, 1024, 0, stream>>>(imp1, imp2, m0w, m1w, m2w,
                                     idx0, idx1, idx2, cnts,
                                     out + o5, out + o6, out + o7,
                                     out + o8, out + o9, out + o10,
                                     out + o11, out + o12, out + o13);

  // bulk data movement (HBM-bound; b128 everywhere)
  pt_g0<<<N0, 192, 0, stream>>>(img, idx0, cnts + 0, out + o0);
  pt_g1<<<N1, 192, 0, stream>>>(img, idx1, cnts + 1, out + o1);
  pt_g2<<<N2, 192, 0, stream>>>(img, idx2, cnts + 2, out + o2);
  pt_f0<<<N1, 256, 0, stream>>>(img, idx1, cnts + 1, out + o3);
  pt_f1<<<N2, 256, 0, stream>>>(img, idx2, cnts + 2, out + o4);
}